// PSROIPool_56100862820591
// MI455X (gfx1250) — compile-verified
//
#include <hip/hip_runtime.h>
#include <hip/hip_bf16.h>

// Position-sensitive ROI pooling (R-FCN) for MI455X (gfx1250).
// Per-channel block; stage FM[c] (64x64 f32, 16KB) in LDS via the CDNA5
// async direct-to-LDS path (GLOBAL_LOAD_ASYNC_TO_LDS_B128, ASYNCcnt); per
// 16-ROI tile compute V = FM[c] x Mw^T with V_WMMA_F32_16X16X4_F32
// (fp32-exact, masks built in-registers from ROI boxes), then contract with
// the row mask and normalize. Problem is HBM/L2-bound (17MB read once,
// 2.5 GFLOP), so fp32 WMMA is the right precision: exact math, bandwidth
// unchanged.

typedef float v2f __attribute__((ext_vector_type(2)));
typedef float v8f __attribute__((ext_vector_type(8)));
typedef int   v4i __attribute__((vector_size(16)));   // matches builtin param type

#define N_TARGETS 21
#define R_HW      7
#define FM_HH     64
#define FM_WW     64
#define N_CH      (N_TARGETS * R_HW * R_HW)   // 1029 channels
#define LDS_STRIDE 68                         // 64 + pad (bank-conflict avoidance)
#define VBUF_STRIDE 17                        // 16 + pad
#define TILE_R    16
#define WAVES     4
#define BLOCK_T   (WAVES * 32)

// Address-space casts via integer detour (clang forbids direct AS pointer
// casts). Generic->AS3: flat LDS address truncated to 32 bits == LDS offset
// Generic->AS1 is numerically identity.
#define TO_GLOBAL4(p) ((__attribute__((address_space(1))) v4i*)(unsigned long long)(p))
#define TO_LDS4(p)    ((__attribute__((address_space(3))) v4i*)(unsigned int)(unsigned long long)(p))

__global__ __launch_bounds__(BLOCK_T)
void psroi_wmma_kernel(const float* __restrict__ FM,
                       const float* __restrict__ rois,
                       float* __restrict__ out,
                       int n_rois)
{
    __shared__ float fmLds[FM_HH * LDS_STRIDE];              // 17,408 B
    __shared__ float vbuf[WAVES][FM_HH * VBUF_STRIDE];       // 4 * 4,352 B

    const int c      = blockIdx.x;          // channel: c = t*49 + p*7 + q
    const int q      = c % R_HW;
    const int p      = (c / R_HW) % R_HW;
    const int tid    = threadIdx.x;
    const int lane   = tid & 31;
    const int wv     = tid >> 5;
    const int half   = lane >> 4;           // K-half select per WMMA A/B layout
    const int lane16 = lane & 15;

    __builtin_prefetch(rois, 0, 3);         // global_prefetch_b8 (CDNA5)

    // ---- Stage FM channel into LDS (128 threads x 32 floats) ----
    {
        const float* src = FM + (size_t)c * (FM_HH * FM_WW);
        const int row  = tid >> 1;          // 0..63
        const int hseg = tid & 1;           // which 32-float half-row
        const float* g = src + row * FM_WW + hseg * 32;
        float* dst = fmLds + row * LDS_STRIDE + hseg * 32;
#if __has_builtin(__builtin_amdgcn_global_load_async_to_lds_b128)
        // Async direct-to-LDS copy: no VGPR round-trip, tracked by ASYNCcnt.
#pragma unroll
        for (int v = 0; v < 8; ++v) {
            __builtin_amdgcn_global_load_async_to_lds_b128(
                TO_GLOBAL4(g + 4 * v), TO_LDS4(dst + 4 * v), 0, 0);
        }
#if __has_builtin(__builtin_amdgcn_s_wait_asynccnt)
        __builtin_amdgcn_s_wait_asynccnt(0);
#else
        asm volatile("s_wait_asynccnt 0x0" ::: "memory");
#endif
#else
        const float4* s4 = (const float4*)g;
#pragma unroll
        for (int v = 0; v < 8; ++v) {
            float4 d = s4[v];
            dst[4*v+0] = d.x; dst[4*v+1] = d.y;
            dst[4*v+2] = d.z; dst[4*v+3] = d.w;
        }
#endif
    }
    __syncthreads();

    const int n_tiles = (n_rois + TILE_R - 1) / TILE_R;

    for (int tile = wv; tile < n_tiles; tile += WAVES) {
        // ---- per-lane ROI box -> bin [lo,hi) ranges (lane16 = column n) ----
        const int r = tile * TILE_R + lane16;
        const bool valid = (r < n_rois);
        float x1 = 0.f, y1 = 0.f, x2 = 0.f, y2 = 0.f;
        if (valid) {
            x1 = rois[4*r+0]; y1 = rois[4*r+1];
            x2 = rois[4*r+2]; y2 = rois[4*r+3];
        }
        const float bw = fmaxf(x2 - x1, 0.1f) * (1.0f / R_HW);
        const float bh = fmaxf(y2 - y1, 0.1f) * (1.0f / R_HW);
        int lo_w = (int)fminf(fmaxf(floorf(x1 +  q        * bw), 0.f), (float)FM_WW);
        int hi_w = (int)fminf(fmaxf(ceilf (x1 + (q + 1.f) * bw), 0.f), (float)FM_WW);
        int lo_h = (int)fminf(fmaxf(floorf(y1 +  p        * bh), 0.f), (float)FM_HH);
        int hi_h = (int)fminf(fmaxf(ceilf (y1 + (p + 1.f) * bh), 0.f), (float)FM_HH);
        if (!valid) { lo_w = hi_w = 0; lo_h = hi_h = 0; }

        // ---- Stage 1: V(64h x 16r) = FM[c] (64x64) x Mw^T (64x16) ----
        v8f acc0 = {}, acc1 = {}, acc2 = {}, acc3 = {};
#pragma unroll
        for (int k = 0; k < 16; ++k) {
            const int w0 = 4 * k + 2 * half;     // K element pair per lane half
            v2f b;                               // B[kk][n]: column mask of roi n
            b.x = (w0     >= lo_w && w0     < hi_w) ? 1.0f : 0.0f;
            b.y = (w0 + 1 >= lo_w && w0 + 1 < hi_w) ? 1.0f : 0.0f;
            const float* col = fmLds + w0;       // A[m][kk] = FM[c][m][w0+kk]
            v2f a0 = *(const v2f*)(col + (0*16 + lane16) * LDS_STRIDE);
            v2f a1 = *(const v2f*)(col + (1*16 + lane16) * LDS_STRIDE);
            v2f a2 = *(const v2f*)(col + (2*16 + lane16) * LDS_STRIDE);
            v2f a3 = *(const v2f*)(col + (3*16 + lane16) * LDS_STRIDE);
            acc0 = __builtin_amdgcn_wmma_f32_16x16x4_f32(false, a0, false, b, (short)0, acc0, false, false);
            acc1 = __builtin_amdgcn_wmma_f32_16x16x4_f32(false, a1, false, b, (short)0, acc1, false, false);
            acc2 = __builtin_amdgcn_wmma_f32_16x16x4_f32(false, a2, false, b, (short)0, acc2, false, false);
            acc3 = __builtin_amdgcn_wmma_f32_16x16x4_f32(false, a3, false, b, (short)0, acc3, false, false);
        }

        // ---- Spill V tiles to per-wave LDS (C/D layout: vgpr j -> row j+8*half) ----
        float* vb = &vbuf[wv][0];
#pragma unroll
        for (int j = 0; j < 8; ++j) {
            const int rw = j + 8 * half;
            vb[(0*16 + rw) * VBUF_STRIDE + lane16] = acc0[j];
            vb[(1*16 + rw) * VBUF_STRIDE + lane16] = acc1[j];
            vb[(2*16 + rw) * VBUF_STRIDE + lane16] = acc2[j];
            vb[(3*16 + rw) * VBUF_STRIDE + lane16] = acc3[j];
        }
        __builtin_amdgcn_wave_barrier();   // per-wave buffer; DScnt deps tracked

        // ---- Stage 2: s = sum_{h in [lo_h,hi_h)} V[h][n], normalize, store ----
        if (lane < 16 && valid) {
            float s = 0.0f;
            for (int h = lo_h; h < hi_h; ++h)
                s += vb[h * VBUF_STRIDE + lane16];
            const float cnt = (float)((hi_h - lo_h) * (hi_w - lo_w));
            // out[r, t, p, q] flat = r*1029 + c  (since c = t*49 + p*7 + q)
            out[(size_t)r * N_CH + c] = s / fmaxf(cnt, 1.0f);
        }
        __builtin_amdgcn_wave_barrier();
    }
}

extern "C" void kernel_launch(void* const* d_in, const int* in_sizes, int n_in,
                              void* d_out, int out_size, void* d_ws, size_t ws_size,
                              hipStream_t stream) {
    (void)n_in; (void)out_size; (void)d_ws; (void)ws_size;
    const float* FM   = (const float*)d_in[0];
    const float* rois = (const float*)d_in[1];
    const int n_rois  = in_sizes[1] / 4;     // rois is [R,4]
    float* out = (float*)d_out;

    dim3 grid(N_CH);        // 1029 blocks: one per PS channel
    dim3 block(BLOCK_T);    // 4 wave32's
    psroi_wmma_kernel<<<grid, block, 0, stream>>>(FM, rois, out, n_rois);
}